// S5SSM_75428215652396
// MI455X (gfx1250) — compile-verified
//
#include <hip/hip_runtime.h>
#include <cmath>

// S5 SSM fused: Bu GEMM -> chunked diagonal scan -> readout GEMM.
// fp32 WMMA (v_wmma_f32_16x16x4_f32): HBM-bound (~192MB, ~35 GFLOP), so full
// fp32 matrix precision is free. Signal tiles staged memory->LDS with gfx1250
// async loads (ASYNCcnt). One A-fragment ds_load_b64 set feeds 16 WMMAs.

typedef float v8f __attribute__((ext_vector_type(8)));
typedef float v2f __attribute__((ext_vector_type(2)));

#define BATCH   8
#define LSEQ    8192
#define HDIM    256
#define PDIM    256
#define CH      64            // timesteps per chunk
#define NC      (LSEQ / CH)   // 128 chunks per batch
#define LOG2CH  6
#define THREADS 256

#define USE_ASYNC_STAGE 1

// ---- workspace layout (float offsets). Needs ~5.3 MB. ----
#define OFF_WBRE   0
#define OFF_WBIM   (OFF_WBRE  + HDIM * PDIM)
#define OFF_WCRE   (OFF_WBIM  + HDIM * PDIM)
#define OFF_WCIMN  (OFF_WCRE  + PDIM * HDIM)
#define OFF_LAMRE  (OFF_WCIMN + PDIM * HDIM)
#define OFF_LAMIM  (OFF_LAMRE + PDIM)
#define OFF_LCHRE  (OFF_LAMIM + PDIM)
#define OFF_LCHIM  (OFF_LCHRE + PDIM)
#define OFF_ERE    (OFF_LCHIM + PDIM)
#define OFF_EIM    (OFF_ERE   + BATCH * NC * PDIM)
#define OFF_CRE    (OFF_EIM   + BATCH * NC * PDIM)
#define OFF_CIM    (OFF_CRE   + BATCH * NC * PDIM)

__device__ __forceinline__ v8f wmma_f32(v2f a, v2f b, v8f c) {
  // 8 args: (neg_a, A, neg_b, B, c_mod, C, reuse_a, reuse_b)
  return __builtin_amdgcn_wmma_f32_16x16x4_f32(false, a, false, b, (short)0, c,
                                               false, false);
}

// ---------------------------------------------------------------------------
// k0: ZOH discretization + weight transposes.
// ---------------------------------------------------------------------------
__global__ void k0_prep(const float* __restrict__ Lre, const float* __restrict__ Lim,
                        const float* __restrict__ Bri, const float* __restrict__ Cre,
                        const float* __restrict__ Cim, const float* __restrict__ lstep,
                        float* __restrict__ ws) {
  int p = blockIdx.x;   // 0..P-1
  int h = threadIdx.x;  // 0..H-1
  float lr = Lre[p], li = Lim[p];
  float dt = __expf(lstep[p]);
  float ea = __expf(lr * dt);
  float lbr = ea * __cosf(li * dt);
  float lbi = ea * __sinf(li * dt);
  // F = (Lambda_bar - 1) / Lambda
  float nr = lbr - 1.0f, ni = lbi;
  float inv = 1.0f / (lr * lr + li * li);
  float fr = (nr * lr + ni * li) * inv;
  float fi = (ni * lr - nr * li) * inv;
  float btr = Bri[((size_t)p * HDIM + h) * 2 + 0];
  float bti = Bri[((size_t)p * HDIM + h) * 2 + 1];
  ws[OFF_WBRE  + h * PDIM + p] = fr * btr - fi * bti;
  ws[OFF_WBIM  + h * PDIM + p] = fr * bti + fi * btr;
  ws[OFF_WCRE  + p * HDIM + h] =  Cre[h * PDIM + p];
  ws[OFF_WCIMN + p * HDIM + h] = -Cim[h * PDIM + p];
  if (h == 0) {
    ws[OFF_LAMRE + p] = lbr;
    ws[OFF_LAMIM + p] = lbi;
    float cr = lbr, ci = lbi;
#pragma unroll
    for (int i = 0; i < LOG2CH; ++i) {   // Lambda_bar^(2^LOG2CH)
      float t = cr * cr - ci * ci;
      ci = 2.0f * cr * ci;
      cr = t;
    }
    ws[OFF_LCHRE + p] = cr;
    ws[OFF_LCHIM + p] = ci;
  }
}

// ---------------------------------------------------------------------------
// Signal tile staging: global -> LDS (ASYNCcnt-tracked direct path)
// ---------------------------------------------------------------------------
__device__ __forceinline__ void stage_signal(const float* __restrict__ sig,
                                             float* __restrict__ Uld,
                                             int b, int t0) {
  const float* src = sig + ((size_t)b * LSEQ + t0) * HDIM;
#if USE_ASYNC_STAGE
  // CH*HDIM*4B = 64KB; 256 threads x 16B = 4KB per sweep -> 16 sweeps.
#pragma unroll
  for (int i = 0; i < (CH * HDIM * 4) / (THREADS * 16); ++i) {
    unsigned int eoff = (unsigned int)((i * THREADS + threadIdx.x) * 16);
    unsigned int ldsaddr =
        (unsigned int)(unsigned long long)(Uld) + eoff;  // addr[31:0] = LDS byte addr
    unsigned long long gaddr = (unsigned long long)(src) + eoff;
    asm volatile("global_load_async_to_lds_b128 %0, %1, off"
                 :
                 : "v"(ldsaddr), "v"(gaddr)
                 : "memory");
  }
  asm volatile("s_wait_asynccnt 0x0" ::: "memory");
#else
  const float4* s4 = (const float4*)src;
  float4* d4 = (float4*)Uld;
#pragma unroll
  for (int i = threadIdx.x; i < (CH * HDIM) / 4; i += THREADS)
    d4[i] = s4[i];
#endif
}

// ---------------------------------------------------------------------------
// Bu(re/im)[CH x P] = U[CH x H] @ Wb(re/im)[H x P] via f32 WMMA 16x16x4.
// Wave owns 4 column groups (re/col0, re/col1, im/col0, im/col1) x 4 row
// tiles = 16 accumulators. Per K-step: 4 ds_load_b64 (A, shared by all 16
// WMMAs) + 4 double-buffered B fragments (each feeds 4 WMMAs).
// ---------------------------------------------------------------------------
__device__ __forceinline__ void gemm_bu(const float* __restrict__ WbRe,
                                        const float* __restrict__ WbIm,
                                        const float* __restrict__ Uld,
                                        float* __restrict__ BuRe,
                                        float* __restrict__ BuIm,
                                        int wave, int lane) {
  const int lrow  = lane & 15;
  const int khalf = (lane >> 4) * 2;
  const int mbase = (lane >> 4) << 3;
  const int col0  = wave * 16 + lrow;
  const int col1  = col0 + 128;

  const float* u0 = Uld + (     lrow) * HDIM + khalf;
  const float* u1 = Uld + (16 + lrow) * HDIM + khalf;
  const float* u2 = Uld + (32 + lrow) * HDIM + khalf;
  const float* u3 = Uld + (48 + lrow) * HDIM + khalf;

  const float* wp[4] = {WbRe + khalf * PDIM + col0, WbRe + khalf * PDIM + col1,
                        WbIm + khalf * PDIM + col0, WbIm + khalf * PDIM + col1};

  v8f acc[4][4] = {};
  v2f b[4], bn[4];
#pragma unroll
  for (int g = 0; g < 4; ++g) {
    b[g].x = wp[g][0];
    b[g].y = wp[g][PDIM];
  }

#pragma unroll 2
  for (int k = 0; k < HDIM; k += 4) {
    const int kn = ((k + 4) & (HDIM - 1)) * PDIM;   // wraps to row 0 on last iter
#pragma unroll
    for (int g = 0; g < 4; ++g) {
      bn[g].x = wp[g][kn];
      bn[g].y = wp[g][kn + PDIM];
    }
    v2f a0 = *(const v2f*)(u0 + k);
    v2f a1 = *(const v2f*)(u1 + k);
    v2f a2 = *(const v2f*)(u2 + k);
    v2f a3 = *(const v2f*)(u3 + k);
#pragma unroll
    for (int g = 0; g < 4; ++g) {
      acc[g][0] = wmma_f32(a0, b[g], acc[g][0]);
      acc[g][1] = wmma_f32(a1, b[g], acc[g][1]);
      acc[g][2] = wmma_f32(a2, b[g], acc[g][2]);
      acc[g][3] = wmma_f32(a3, b[g], acc[g][3]);
    }
#pragma unroll
    for (int g = 0; g < 4; ++g) b[g] = bn[g];
  }

  float* op[4] = {BuRe + col0, BuRe + col1, BuIm + col0, BuIm + col1};
#pragma unroll
  for (int g = 0; g < 4; ++g)
#pragma unroll
    for (int mt = 0; mt < 4; ++mt)
#pragma unroll
      for (int i = 0; i < 8; ++i)
        op[g][(mt * 16 + mbase + i) * PDIM] = acc[g][mt][i];   // M = i + 8*(lane>=16)
}

// ---------------------------------------------------------------------------
// One readout pass: acc += X[CH x P] @ W[P x H] restricted to this wave's two
// 16-wide column groups. Called twice (re, im) on the same accumulators.
// ---------------------------------------------------------------------------
__device__ __forceinline__ void readout_pass(const float* __restrict__ W,
                                             const float* __restrict__ X,
                                             int col0, int col1, int lrow,
                                             int khalf, v8f acc[2][4]) {
  const float* w0 = W + khalf * HDIM + col0;
  const float* w1 = W + khalf * HDIM + col1;
  const float* x0 = X + (     lrow) * PDIM + khalf;
  const float* x1 = X + (16 + lrow) * PDIM + khalf;
  const float* x2 = X + (32 + lrow) * PDIM + khalf;
  const float* x3 = X + (48 + lrow) * PDIM + khalf;
  v2f b0, b1, bn0, bn1;
  b0.x = w0[0]; b0.y = w0[HDIM];
  b1.x = w1[0]; b1.y = w1[HDIM];
#pragma unroll 2
  for (int k = 0; k < PDIM; k += 4) {
    const int kn = ((k + 4) & (PDIM - 1)) * HDIM;
    bn0.x = w0[kn]; bn0.y = w0[kn + HDIM];
    bn1.x = w1[kn]; bn1.y = w1[kn + HDIM];
    v2f a0 = *(const v2f*)(x0 + k);
    v2f a1 = *(const v2f*)(x1 + k);
    v2f a2 = *(const v2f*)(x2 + k);
    v2f a3 = *(const v2f*)(x3 + k);
    acc[0][0] = wmma_f32(a0, b0, acc[0][0]);
    acc[0][1] = wmma_f32(a1, b0, acc[0][1]);
    acc[0][2] = wmma_f32(a2, b0, acc[0][2]);
    acc[0][3] = wmma_f32(a3, b0, acc[0][3]);
    acc[1][0] = wmma_f32(a0, b1, acc[1][0]);
    acc[1][1] = wmma_f32(a1, b1, acc[1][1]);
    acc[1][2] = wmma_f32(a2, b1, acc[1][2]);
    acc[1][3] = wmma_f32(a3, b1, acc[1][3]);
    b0 = bn0; b1 = bn1;
  }
}

// ---------------------------------------------------------------------------
// k1: per-chunk local end state e = scan(Bu) from zero.
// ---------------------------------------------------------------------------
__global__ void __launch_bounds__(THREADS, 1)
k1_chunk_ends(const float* __restrict__ sig, float* __restrict__ ws) {
  extern __shared__ float smem[];
  float* Uld  = smem;                       // CH*HDIM
  float* BuRe = smem + CH * HDIM;           // CH*PDIM
  float* BuIm = BuRe + CH * PDIM;           // CH*PDIM
  int blk = blockIdx.x;                     // b*NC + chunk
  int b = blk / NC, ck = blk - b * NC;
  int t0 = ck * CH;
  int lane = threadIdx.x & 31, wave = threadIdx.x >> 5;

  for (int i = threadIdx.x; i < (HDIM * PDIM) / 32; i += THREADS) {
    __builtin_prefetch(ws + OFF_WBRE + i * 32, 0, 0);
    __builtin_prefetch(ws + OFF_WBIM + i * 32, 0, 0);
  }
  stage_signal(sig, Uld, b, t0);
  __syncthreads();
  gemm_bu(ws + OFF_WBRE, ws + OFF_WBIM, Uld, BuRe, BuIm, wave, lane);
  __syncthreads();

  int p = threadIdx.x;
  float lr = ws[OFF_LAMRE + p], li = ws[OFF_LAMIM + p];
  const float* br_p = BuRe + p;
  const float* bi_p = BuIm + p;
  float sr = 0.0f, si = 0.0f;
#pragma unroll 4
  for (int t = 0; t < CH; ++t) {
    float br = br_p[t * PDIM];
    float bi = bi_p[t * PDIM];
    float nr = lr * sr - li * si + br;
    float ni = lr * si + li * sr + bi;
    sr = nr; si = ni;
  }
  ws[OFF_ERE + (size_t)blk * PDIM + p] = sr;
  ws[OFF_EIM + (size_t)blk * PDIM + p] = si;
}

// ---------------------------------------------------------------------------
// k1b: per-batch scan of chunk end states -> carry-in for every chunk.
// ---------------------------------------------------------------------------
__global__ void k1b_scan_carries(float* __restrict__ ws) {
  int b = blockIdx.x, p = threadIdx.x;
  float lr = ws[OFF_LCHRE + p], li = ws[OFF_LCHIM + p];
  float gr = 0.0f, gi = 0.0f;
  for (int j = 0; j < NC; ++j) {
    size_t idx = ((size_t)b * NC + j) * PDIM + p;
    ws[OFF_CRE + idx] = gr;
    ws[OFF_CIM + idx] = gi;
    float er = ws[OFF_ERE + idx], ei = ws[OFF_EIM + idx];
    float nr = lr * gr - li * gi + er;
    float ni = lr * gi + li * gr + ei;
    gr = nr; gi = ni;
  }
}

// ---------------------------------------------------------------------------
// k2: recompute Bu, scan with carry (overwrite LDS with x_t), readout GEMM:
//   y = X_re @ WcRe + X_im @ WcImNeg + D*u
// ---------------------------------------------------------------------------
__global__ void __launch_bounds__(THREADS, 1)
k2_final(const float* __restrict__ sig, const float* __restrict__ Dvec,
         float* __restrict__ y, float* __restrict__ ws) {
  extern __shared__ float smem[];
  float* Uld  = smem;
  float* BuRe = smem + CH * HDIM;
  float* BuIm = BuRe + CH * PDIM;
  int blk = blockIdx.x;
  int b = blk / NC, ck = blk - b * NC;
  int t0 = ck * CH;
  int lane = threadIdx.x & 31, wave = threadIdx.x >> 5;

  for (int i = threadIdx.x; i < (HDIM * PDIM) / 32; i += THREADS) {
    __builtin_prefetch(ws + OFF_WCRE  + i * 32, 0, 0);
    __builtin_prefetch(ws + OFF_WCIMN + i * 32, 0, 0);
  }
  stage_signal(sig, Uld, b, t0);
  __syncthreads();
  gemm_bu(ws + OFF_WBRE, ws + OFF_WBIM, Uld, BuRe, BuIm, wave, lane);
  __syncthreads();

  // inclusive scan seeded by carry; overwrite Bu with x_t
  {
    int p = threadIdx.x;
    size_t cidx = (size_t)blk * PDIM + p;
    float sr = ws[OFF_CRE + cidx], si = ws[OFF_CIM + cidx];
    float lr = ws[OFF_LAMRE + p],  li = ws[OFF_LAMIM + p];
    float* br_p = BuRe + p;
    float* bi_p = BuIm + p;
#pragma unroll 4
    for (int t = 0; t < CH; ++t) {
      float br = br_p[t * PDIM];
      float bi = bi_p[t * PDIM];
      float nr = lr * sr - li * si + br;
      float ni = lr * si + li * sr + bi;
      sr = nr; si = ni;
      br_p[t * PDIM] = sr;
      bi_p[t * PDIM] = si;
    }
  }
  __syncthreads();

  // readout GEMM: wave owns h-columns [16w,16w+16) and [16w+128,16w+144).
  const int lrow  = lane & 15;
  const int khalf = (lane >> 4) * 2;
  const int mbase = (lane >> 4) << 3;
  const int col0  = wave * 16 + lrow;
  const int col1  = col0 + 128;

  v8f acc[2][4] = {};
  readout_pass(ws + OFF_WCRE,  BuRe, col0, col1, lrow, khalf, acc);  // +X_re@C_re
  readout_pass(ws + OFF_WCIMN, BuIm, col0, col1, lrow, khalf, acc);  // -X_im@C_im

  // epilogue: + D*u, direct store from accumulator layout
  const int cols[2] = {col0, col1};
#pragma unroll
  for (int c = 0; c < 2; ++c) {
    const int h = cols[c];
    const float d = Dvec[h];
#pragma unroll
    for (int mt = 0; mt < 4; ++mt) {
      const int mrow = mt * 16 + mbase;
#pragma unroll
      for (int i = 0; i < 8; ++i) {
        const int m = mrow + i;
        y[((size_t)(b * LSEQ + t0 + m)) * HDIM + h] =
            acc[c][mt][i] + d * Uld[m * HDIM + h];
      }
    }
  }
}

// ---------------------------------------------------------------------------
extern "C" void kernel_launch(void* const* d_in, const int* in_sizes, int n_in,
                              void* d_out, int out_size, void* d_ws, size_t ws_size,
                              hipStream_t stream) {
  const float* sig = (const float*)d_in[0];
  const float* Lre = (const float*)d_in[1];
  const float* Lim = (const float*)d_in[2];
  const float* Bri = (const float*)d_in[3];
  const float* Cre = (const float*)d_in[4];
  const float* Cim = (const float*)d_in[5];
  const float* Dv  = (const float*)d_in[6];
  const float* ls  = (const float*)d_in[7];
  float* y  = (float*)d_out;
  float* ws = (float*)d_ws;

  const size_t smem = (size_t)(CH * HDIM + 2 * CH * PDIM) * sizeof(float); // 192 KB
  (void)hipFuncSetAttribute((const void*)k1_chunk_ends,
                            hipFuncAttributeMaxDynamicSharedMemorySize, (int)smem);
  (void)hipFuncSetAttribute((const void*)k2_final,
                            hipFuncAttributeMaxDynamicSharedMemorySize, (int)smem);

  k0_prep<<<PDIM, HDIM, 0, stream>>>(Lre, Lim, Bri, Cre, Cim, ls, ws);
  k1_chunk_ends<<<BATCH * NC, THREADS, smem, stream>>>(sig, ws);
  k1b_scan_carries<<<BATCH, PDIM, 0, stream>>>(ws);
  k2_final<<<BATCH * NC, THREADS, smem, stream>>>(sig, Dv, y, ws);
}